// IpaScore_2430951489757
// MI455X (gfx1250) — compile-verified
//
#include <hip/hip_runtime.h>
#include <cmath>

#define NRES 384
#define ROWS 768          // B*N
#define NBATCH 2
#define CSd  256
#define CZd  128
#define Hd   8
#define CHd  256
#define PQd  8
#define PVd  12
#define THd  4
#define SCALE_F 0.1f

typedef __attribute__((ext_vector_type(2))) float v2f;
typedef __attribute__((ext_vector_type(8))) float v8f;

// ---------------------------------------------------------------------------
// Generic batched WMMA GEMM:  C[z] = act(alpha * A[z] @ B[z] + bias)
// Batch z decomposed as (z/innerDiv, z%innerDiv) with independent strides.
// BT=1 means B stored as [N][K] (computes A @ B^T).
// Ragged M / Nc handled by CLAMPING indices (no exec-mask branches in the
// inner loop): out-of-range rows/cols read row/col 0 — their products land
// only in output elements that are never stored, so results are unaffected.
// Each wave computes a 16x32 output slab (two 16x16 WMMA tiles sharing the
// A fragment).  EXEC is all-ones at every WMMA.
// ---------------------------------------------------------------------------
template <int BT>
__launch_bounds__(64)
__global__ void gemm_wmma_kernel(const float* __restrict__ A, int lda, long long aO, long long aI,
                                 const float* __restrict__ Bm, int ldb, long long bO, long long bI,
                                 const float* __restrict__ bias,
                                 float* __restrict__ C, int ldc, long long cO, long long cI,
                                 int M, int Nc, int K, int innerDiv, float alpha, int relu)
{
    int z  = blockIdx.z;
    int zo = z / innerDiv, zi = z % innerDiv;
    const float* Ab = A  + zo * aO + zi * aI;
    const float* Bb = Bm + zo * bO + zi * bI;
    float*       Cb = C  + zo * cO + zi * cI;

    int wave = threadIdx.x >> 5;
    int lane = threadIdx.x & 31;
    int m0 = blockIdx.x * 16;
    int n0 = (blockIdx.y * 2 + wave) * 32;
    if (n0 >= Nc) return;                        // uniform within wave

    int half = lane >> 4;                        // 0 or 1
    int l15  = lane & 15;
    int arow  = m0 + l15;
    int ar    = (arow < M) ? arow : 0;           // clamped (safe, never stored)
    int bcol0 = n0 + l15;
    int bcol1 = n0 + 16 + l15;
    int bc0 = (bcol0 < Nc) ? bcol0 : 0;
    int bc1 = (bcol1 < Nc) ? bcol1 : 0;
    int kk = half * 2;                           // K sub-offset: 0 or 2

    const float* ap = Ab + (long long)ar * lda + kk;

    v8f acc0 = {0.f, 0.f, 0.f, 0.f, 0.f, 0.f, 0.f, 0.f};
    v8f acc1 = {0.f, 0.f, 0.f, 0.f, 0.f, 0.f, 0.f, 0.f};

#pragma unroll 2
    for (int k0 = 0; k0 < K; k0 += 4) {
        v2f a, b0, b1;
        a.x = ap[k0];
        a.y = ap[k0 + 1];
        if (BT) {
            const float* bp0 = Bb + (long long)bc0 * ldb + k0 + kk;
            const float* bp1 = Bb + (long long)bc1 * ldb + k0 + kk;
            b0.x = bp0[0]; b0.y = bp0[1];
            b1.x = bp1[0]; b1.y = bp1[1];
        } else {
            const float* bp = Bb + (long long)(k0 + kk) * ldb;
            b0.x = bp[bc0]; b0.y = bp[bc0 + ldb];
            b1.x = bp[bc1]; b1.y = bp[bc1 + ldb];
        }
        acc0 = __builtin_amdgcn_wmma_f32_16x16x4_f32(false, a, false, b0,
                                                     (short)0, acc0, false, false);
        acc1 = __builtin_amdgcn_wmma_f32_16x16x4_f32(false, a, false, b1,
                                                     (short)0, acc1, false, false);
    }

    float bv0 = bias ? bias[bc0] : 0.f;
    float bv1 = bias ? bias[bc1] : 0.f;
    int crow0 = m0 + half * 8;
#pragma unroll
    for (int r = 0; r < 8; ++r) {
        int crow = crow0 + r;
        if (crow < M) {
            if (bcol0 < Nc) {
                float v = acc0[r] * alpha + bv0;
                if (relu) v = fmaxf(v, 0.f);
                Cb[(long long)crow * ldc + bcol0] = v;
            }
            if (bcol1 < Nc) {
                float v = acc1[r] * alpha + bv1;
                if (relu) v = fmaxf(v, 0.f);
                Cb[(long long)crow * ldc + bcol1] = v;
            }
        }
    }
}

// ---------------------------------------------------------------------------
// LayerNorm over 256 columns, one workgroup (256 threads) per row.
// ---------------------------------------------------------------------------
__global__ void ln_kernel(const float* __restrict__ src, float* __restrict__ dst,
                          const float* __restrict__ g, const float* __restrict__ bta)
{
    __shared__ float red[256];
    int row = blockIdx.x, tid = threadIdx.x;
    float v = src[(size_t)row * CSd + tid];
    red[tid] = v; __syncthreads();
    for (int s = 128; s > 0; s >>= 1) { if (tid < s) red[tid] += red[tid + s]; __syncthreads(); }
    float mean = red[0] * (1.0f / CSd); __syncthreads();
    float d = v - mean;
    red[tid] = d * d; __syncthreads();
    for (int s = 128; s > 0; s >>= 1) { if (tid < s) red[tid] += red[tid + s]; __syncthreads(); }
    float var = red[0] * (1.0f / CSd);
    dst[(size_t)row * CSd + tid] = d * rsqrtf(var + 1e-5f) * g[tid] + bta[tid];
}

// ---------------------------------------------------------------------------
// Softmax over last dim (384) with mask bias, 128 threads/row.
// ---------------------------------------------------------------------------
__global__ void softmax_kernel(float* __restrict__ att, const float* __restrict__ mask,
                               int rowsPerB)
{
    __shared__ float red[128];
    int row = blockIdx.x, tid = threadIdx.x;
    int b = row / rowsPerB;
    float vals[3];
    float mx = -1e30f;
#pragma unroll
    for (int i = 0; i < 3; ++i) {
        int j = tid + i * 128;
        float v = att[(long long)row * NRES + j] + (mask[b * NRES + j] - 1.0f) * 1e9f;
        vals[i] = v; mx = fmaxf(mx, v);
    }
    red[tid] = mx; __syncthreads();
    for (int s = 64; s > 0; s >>= 1) { if (tid < s) red[tid] = fmaxf(red[tid], red[tid + s]); __syncthreads(); }
    mx = red[0]; __syncthreads();
    float sum = 0.f;
#pragma unroll
    for (int i = 0; i < 3; ++i) { vals[i] = expf(vals[i] - mx); sum += vals[i]; }
    red[tid] = sum; __syncthreads();
    for (int s = 64; s > 0; s >>= 1) { if (tid < s) red[tid] += red[tid + s]; __syncthreads(); }
    float inv = 1.0f / red[0];
#pragma unroll
    for (int i = 0; i < 3; ++i)
        att[(long long)row * NRES + tid + i * 128] = vals[i] * inv;
}

// ---------------------------------------------------------------------------
// IPA logit bias: att[b,h,i,j] += c_b * (edge[b,i,j,:] . wb[:,h])
//                              + c_pt*softplus(gamma[h]) * d2[b,i,j,h]
// One thread per (b,i,j).
// ---------------------------------------------------------------------------
__global__ void ipa_bias_kernel(float* __restrict__ att, const float* __restrict__ edge,
                                const float* __restrict__ wb, const float* __restrict__ gamma,
                                const float* __restrict__ qp, const float* __restrict__ kp,
                                float c_b, float c_pt)
{
    __shared__ float wbs[CZd * Hd];
    __shared__ float hw[Hd];
    int t = threadIdx.x;
    for (int i = t; i < CZd * Hd; i += 256) wbs[i] = wb[i];
    if (t < Hd) {
        float g = gamma[t];
        hw[t] = c_pt * ((g > 20.f) ? g : log1pf(expf(g)));
    }
    __syncthreads();

    long long idx = (long long)blockIdx.x * 256 + t;     // over B*N*N
    int j = (int)(idx % NRES);
    long long tmp = idx / NRES;
    int i = (int)(tmp % NRES);
    int b = (int)(tmp / NRES);

    float acc[Hd] = {0,0,0,0,0,0,0,0};
    const float* ep = edge + idx * CZd;
    for (int c = 0; c < CZd; c += 4) {
        float4 e = *(const float4*)(ep + c);
#pragma unroll
        for (int h = 0; h < Hd; ++h)
            acc[h] += e.x * wbs[(c + 0) * Hd + h] + e.y * wbs[(c + 1) * Hd + h]
                    + e.z * wbs[(c + 2) * Hd + h] + e.w * wbs[(c + 3) * Hd + h];
    }
    const float* qpr = qp + ((long long)b * NRES + i) * (Hd * PQd * 3);
    const float* kpr = kp + ((long long)b * NRES + j) * (Hd * PQd * 3);
#pragma unroll
    for (int h = 0; h < Hd; ++h) {
        float d2 = 0.f;
#pragma unroll
        for (int p = 0; p < PQd; ++p) {
            int o = h * (PQd * 3) + p * 3;
            float dx = qpr[o] - kpr[o];
            float dy = qpr[o + 1] - kpr[o + 1];
            float dz = qpr[o + 2] - kpr[o + 2];
            d2 += dx * dx + dy * dy + dz * dz;
        }
        att[(((long long)b * Hd + h) * NRES + i) * NRES + j] += c_b * acc[h] + hw[h] * d2;
    }
}

// ---------------------------------------------------------------------------
// Rigid-frame helpers
// ---------------------------------------------------------------------------
__global__ void init_s_kernel(const float* __restrict__ in_node, const float* __restrict__ resm,
                              float* __restrict__ s)
{
    long long idx = (long long)blockIdx.x * 256 + threadIdx.x;
    s[idx] = in_node[idx] * resm[idx / CSd];
}

__global__ void init_rigid_kernel(const float* __restrict__ rig, const float* __restrict__ resm,
                                  const float* __restrict__ fixm,
                                  float* __restrict__ R, float* __restrict__ t, float* __restrict__ dm)
{
    int n = blockIdx.x * blockDim.x + threadIdx.x;
    if (n >= ROWS) return;
    const float* q = rig + n * 7;
    float w = q[0], x = q[1], y = q[2], z = q[3];
    float inv = rsqrtf(w * w + x * x + y * y + z * z);
    w *= inv; x *= inv; y *= inv; z *= inv;
    float* Rn = R + n * 9;
    Rn[0] = 1 - 2 * (y * y + z * z); Rn[1] = 2 * (x * y - w * z); Rn[2] = 2 * (x * z + w * y);
    Rn[3] = 2 * (x * y + w * z); Rn[4] = 1 - 2 * (x * x + z * z); Rn[5] = 2 * (y * z - w * x);
    Rn[6] = 2 * (x * z - w * y); Rn[7] = 2 * (y * z + w * x); Rn[8] = 1 - 2 * (x * x + y * y);
    t[n * 3 + 0] = q[4] * SCALE_F;
    t[n * 3 + 1] = q[5] * SCALE_F;
    t[n * 3 + 2] = q[6] * SCALE_F;
    dm[n] = (1.f - fixm[n]) * resm[n];
}

// p' = R p + t (in place), one thread per point
__global__ void apply_rigid_kernel(float* __restrict__ pts, const float* __restrict__ R,
                                   const float* __restrict__ t, int npts)
{
    long long idx = (long long)blockIdx.x * 256 + threadIdx.x;
    int pi = (int)(idx % npts);
    long long n = idx / npts;
    float* p = pts + (n * npts + pi) * 3;
    const float* Rn = R + n * 9;
    const float* tn = t + n * 3;
    float x = p[0], y = p[1], z = p[2];
    p[0] = Rn[0] * x + Rn[1] * y + Rn[2] * z + tn[0];
    p[1] = Rn[3] * x + Rn[4] * y + Rn[5] * z + tn[1];
    p[2] = Rn[6] * x + Rn[7] * y + Rn[8] * z + tn[2];
}

// optl = R^T (opt - t), onrm = |optl|, written into concat buffer
__global__ void point_post_kernel(const float* __restrict__ opt, const float* __restrict__ R,
                                  const float* __restrict__ t, float* __restrict__ ocat)
{
    long long idx = (long long)blockIdx.x * 256 + threadIdx.x;   // ROWS * 96
    int hp = (int)(idx % (Hd * PVd));
    long long n = idx / (Hd * PVd);
    const float* Rn = R + n * 9;
    const float* tn = t + n * 3;
    const float* p = opt + (n * (Hd * PVd) + hp) * 3;
    float x = p[0] - tn[0], y = p[1] - tn[1], z = p[2] - tn[2];
    float lx = Rn[0] * x + Rn[3] * y + Rn[6] * z;
    float ly = Rn[1] * x + Rn[4] * y + Rn[7] * z;
    float lz = Rn[2] * x + Rn[5] * y + Rn[8] * z;
    float* oc = ocat + n * 3456;
    oc[2048 + hp * 3 + 0] = lx;
    oc[2048 + hp * 3 + 1] = ly;
    oc[2048 + hp * 3 + 2] = lz;
    oc[2048 + 288 + hp] = sqrtf(lx * lx + ly * ly + lz * lz + 1e-8f);
}

__global__ void rigid_update_kernel(const float* __restrict__ upd, float* __restrict__ R,
                                    float* __restrict__ t)
{
    int n = blockIdx.x * blockDim.x + threadIdx.x;
    if (n >= ROWS) return;
    const float* u = upd + n * 6;
    float w = 1.f, x = u[0], y = u[1], z = u[2];
    float inv = rsqrtf(w * w + x * x + y * y + z * z);
    w *= inv; x *= inv; y *= inv; z *= inv;
    float Ru[9];
    Ru[0] = 1 - 2 * (y * y + z * z); Ru[1] = 2 * (x * y - w * z); Ru[2] = 2 * (x * z + w * y);
    Ru[3] = 2 * (x * y + w * z); Ru[4] = 1 - 2 * (x * x + z * z); Ru[5] = 2 * (y * z - w * x);
    Ru[6] = 2 * (x * z - w * y); Ru[7] = 2 * (y * z + w * x); Ru[8] = 1 - 2 * (x * x + y * y);
    float* Rn = R + n * 9;
    float r[9];
#pragma unroll
    for (int i = 0; i < 9; ++i) r[i] = Rn[i];
    // t = R * u[3:] + t (old R)
    t[n * 3 + 0] += r[0] * u[3] + r[1] * u[4] + r[2] * u[5];
    t[n * 3 + 1] += r[3] * u[3] + r[4] * u[4] + r[5] * u[5];
    t[n * 3 + 2] += r[6] * u[3] + r[7] * u[4] + r[8] * u[5];
    // R = R * Ru
#pragma unroll
    for (int i = 0; i < 3; ++i)
#pragma unroll
        for (int j = 0; j < 3; ++j)
            Rn[i * 3 + j] = r[i * 3 + 0] * Ru[0 * 3 + j] + r[i * 3 + 1] * Ru[1 * 3 + j] + r[i * 3 + 2] * Ru[2 * 3 + j];
}

// ---------------------------------------------------------------------------
// Small dense heads (Nc = 6 or 2), plain VALU dot products.
// ---------------------------------------------------------------------------
__global__ void small_gemm_kernel(const float* __restrict__ A, const float* __restrict__ W,
                                  const float* __restrict__ bias, const float* __restrict__ mm,
                                  float* __restrict__ C, int K, int Nc)
{
    long long idx = (long long)blockIdx.x * 256 + threadIdx.x;
    if (idx >= (long long)ROWS * Nc) return;
    int c = (int)(idx % Nc);
    long long m = idx / Nc;
    float acc = bias ? bias[c] : 0.f;
    const float* a = A + m * K;
    for (int k = 0; k < K; ++k) acc += a[k] * W[k * Nc + c];
    if (mm) acc *= mm[m];
    C[m * Nc + c] = acc;
}

// ---------------------------------------------------------------------------
// Elementwise helpers
// ---------------------------------------------------------------------------
__global__ void add_inplace_kernel(float* __restrict__ dst, const float* __restrict__ src)
{
    long long i = (long long)blockIdx.x * 256 + threadIdx.x;
    dst[i] += src[i];
}
__global__ void add_out_kernel(float* __restrict__ dst, const float* __restrict__ a,
                               const float* __restrict__ b)
{
    long long i = (long long)blockIdx.x * 256 + threadIdx.x;
    dst[i] = a[i] + b[i];
}
__global__ void add_mask_kernel(float* __restrict__ s, const float* __restrict__ tmp,
                                const float* __restrict__ mask)
{
    long long i = (long long)blockIdx.x * 256 + threadIdx.x;
    s[i] = (s[i] + tmp[i]) * mask[i / CSd];
}

__global__ void assemble_kernel(const float* __restrict__ R, const float* __restrict__ t,
                                const float* __restrict__ u, float* __restrict__ out)
{
    int n = blockIdx.x * blockDim.x + threadIdx.x;
    if (n >= ROWS) return;
    float* o = out + n * 16;
#pragma unroll
    for (int i = 0; i < 9; ++i) o[i] = R[n * 9 + i];
    o[9]  = t[n * 3 + 0] / SCALE_F;
    o[10] = t[n * 3 + 1] / SCALE_F;
    o[11] = t[n * 3 + 2] / SCALE_F;
    float u0 = u[n * 2], u1 = u[n * 2 + 1];
    o[12] = u0; o[13] = u1;
    float inv = rsqrtf(fmaxf(u0 * u0 + u1 * u1, 1e-8f));
    o[14] = u0 * inv; o[15] = u1 * inv;
}

// ---------------------------------------------------------------------------
// Host orchestration
// ---------------------------------------------------------------------------
extern "C" void kernel_launch(void* const* d_in, const int* in_sizes, int n_in,
                              void* d_out, int out_size, void* d_ws, size_t ws_size,
                              hipStream_t stream)
{
    (void)in_sizes; (void)n_in; (void)out_size; (void)ws_size;
    typedef long long ll;
    const float* in_node = (const float*)d_in[0];
    const float* edge    = (const float*)d_in[1];
    const float* resm    = (const float*)d_in[2];
    const float* fixm    = (const float*)d_in[3];
    const float* rig     = (const float*)d_in[4];
    const float* P_ipa_ln_g = (const float*)d_in[5];
    const float* P_ipa_ln_b = (const float*)d_in[6];
    const float* P_wq   = (const float*)d_in[7];
    const float* P_wk   = (const float*)d_in[8];
    const float* P_wv   = (const float*)d_in[9];
    const float* P_wqp  = (const float*)d_in[10];
    const float* P_wkp  = (const float*)d_in[11];
    const float* P_wvp  = (const float*)d_in[12];
    const float* P_wb   = (const float*)d_in[13];
    const float* P_gamma = (const float*)d_in[14];
    const float* P_wo   = (const float*)d_in[15];
    const float* P_tln1g = (const float*)d_in[16];
    const float* P_tln1b = (const float*)d_in[17];
    const float* P_wqkv = (const float*)d_in[18];
    const float* P_bqkv = (const float*)d_in[19];
    const float* P_wao  = (const float*)d_in[20];
    const float* P_bao  = (const float*)d_in[21];
    const float* P_tln2g = (const float*)d_in[22];
    const float* P_tln2b = (const float*)d_in[23];
    const float* P_w1   = (const float*)d_in[24];
    const float* P_b1   = (const float*)d_in[25];
    const float* P_w2   = (const float*)d_in[26];
    const float* P_b2   = (const float*)d_in[27];
    const float* P_postw = (const float*)d_in[28];
    const float* P_trw1 = (const float*)d_in[29];
    const float* P_trb1 = (const float*)d_in[30];
    const float* P_trw2 = (const float*)d_in[31];
    const float* P_trb2 = (const float*)d_in[32];
    const float* P_trw3 = (const float*)d_in[33];
    const float* P_trb3 = (const float*)d_in[34];
    const float* P_trlng = (const float*)d_in[35];
    const float* P_trlnb = (const float*)d_in[36];
    const float* P_bbw  = (const float*)d_in[37];
    const float* P_bbb  = (const float*)d_in[38];
    const float* P_tow1 = (const float*)d_in[39];
    const float* P_tob1 = (const float*)d_in[40];
    const float* P_tow2 = (const float*)d_in[41];
    const float* P_tob2 = (const float*)d_in[42];
    const float* P_towf = (const float*)d_in[43];
    const float* P_tobf = (const float*)d_in[44];

    float* ws = (float*)d_ws;
    size_t off = 0;
    auto alloc = [&](size_t nf) { float* p = ws + off; off += nf; return p; };
    float* sbuf  = alloc((size_t)ROWS * CSd);
    float* xbuf  = alloc((size_t)ROWS * CSd);
    float* xxbuf = alloc((size_t)ROWS * CSd);
    float* thb   = alloc((size_t)ROWS * CSd);
    float* tffb  = alloc((size_t)ROWS * CSd);
    float* qbuf  = alloc((size_t)ROWS * Hd * CHd);
    float* kbuf  = alloc((size_t)ROWS * Hd * CHd);
    float* vbuf  = alloc((size_t)ROWS * Hd * CHd);
    float* qpbuf = alloc((size_t)ROWS * Hd * PQd * 3);
    float* kpbuf = alloc((size_t)ROWS * Hd * PQd * 3);
    float* vpbuf = alloc((size_t)ROWS * Hd * PVd * 3);
    float* optb  = alloc((size_t)ROWS * Hd * PVd * 3);
    float* att   = alloc((size_t)NBATCH * Hd * NRES * NRES);
    float* att2  = alloc((size_t)NBATCH * THd * NRES * NRES);
    float* ocat  = alloc((size_t)ROWS * 3456);
    float* tqkv  = alloc((size_t)ROWS * 768);
    float* Rbuf  = alloc((size_t)ROWS * 9);
    float* tbuf  = alloc((size_t)ROWS * 3);
    float* dmask = alloc((size_t)ROWS);
    float* updb  = alloc((size_t)ROWS * 6);
    float* ubuf  = alloc((size_t)ROWS * 2);

    auto gemm = [&](const float* A, int lda, ll aO, ll aI,
                    const float* Bm, int ldb, ll bO, ll bI, int bT,
                    const float* bias, float* C, int ldc, ll cO, ll cI,
                    int M, int Nc, int K, int batches, int innerDiv, float alpha, int relu) {
        dim3 g((M + 15) / 16, (Nc + 63) / 64, batches);
        if (bT)
            gemm_wmma_kernel<1><<<g, 64, 0, stream>>>(A, lda, aO, aI, Bm, ldb, bO, bI, bias,
                                                      C, ldc, cO, cI, M, Nc, K, innerDiv, alpha, relu);
        else
            gemm_wmma_kernel<0><<<g, 64, 0, stream>>>(A, lda, aO, aI, Bm, ldb, bO, bI, bias,
                                                      C, ldc, cO, cI, M, Nc, K, innerDiv, alpha, relu);
    };

    const float c_qk = (1.0f / sqrtf((float)CHd)) * sqrtf(1.f / 3.f);
    const float c_b  = sqrtf(1.f / 3.f);
    const float c_pt = -0.5f * sqrtf(1.f / (3.f * (PQd * 9.f / 2.f)));

    init_s_kernel<<<ROWS, 256, 0, stream>>>(in_node, resm, sbuf);
    init_rigid_kernel<<<3, 256, 0, stream>>>(rig, resm, fixm, Rbuf, tbuf, dmask);

    for (int nb = 0; nb < 4; ++nb) {
        // ---------------- IPA ----------------
        ln_kernel<<<ROWS, 256, 0, stream>>>(sbuf, xbuf, P_ipa_ln_g + nb * CSd, P_ipa_ln_b + nb * CSd);
        gemm(xbuf, CSd, 0, 0, P_wq + (ll)nb * CSd * 2048, 2048, 0, 0, 0, nullptr,
             qbuf, 2048, 0, 0, ROWS, 2048, CSd, 1, 1, 1.f, 0);
        gemm(xbuf, CSd, 0, 0, P_wk + (ll)nb * CSd * 2048, 2048, 0, 0, 0, nullptr,
             kbuf, 2048, 0, 0, ROWS, 2048, CSd, 1, 1, 1.f, 0);
        gemm(xbuf, CSd, 0, 0, P_wv + (ll)nb * CSd * 2048, 2048, 0, 0, 0, nullptr,
             vbuf, 2048, 0, 0, ROWS, 2048, CSd, 1, 1, 1.f, 0);
        gemm(xbuf, CSd, 0, 0, P_wqp + (ll)nb * CSd * 192, 192, 0, 0, 0, nullptr,
             qpbuf, 192, 0, 0, ROWS, 192, CSd, 1, 1, 1.f, 0);
        gemm(xbuf, CSd, 0, 0, P_wkp + (ll)nb * CSd * 192, 192, 0, 0, 0, nullptr,
             kpbuf, 192, 0, 0, ROWS, 192, CSd, 1, 1, 1.f, 0);
        gemm(xbuf, CSd, 0, 0, P_wvp + (ll)nb * CSd * 288, 288, 0, 0, 0, nullptr,
             vpbuf, 288, 0, 0, ROWS, 288, CSd, 1, 1, 1.f, 0);
        apply_rigid_kernel<<<ROWS * 64 / 256, 256, 0, stream>>>(qpbuf, Rbuf, tbuf, 64);
        apply_rigid_kernel<<<ROWS * 64 / 256, 256, 0, stream>>>(kpbuf, Rbuf, tbuf, 64);
        apply_rigid_kernel<<<ROWS * 96 / 256, 256, 0, stream>>>(vpbuf, Rbuf, tbuf, 96);
        // logits = c_qk * q k^T   (batched over (b,h))
        gemm(qbuf, 2048, (ll)NRES * 2048, CHd, kbuf, 2048, (ll)NRES * 2048, CHd, 1, nullptr,
             att, NRES, (ll)Hd * NRES * NRES, (ll)NRES * NRES,
             NRES, NRES, CHd, NBATCH * Hd, Hd, c_qk, 0);
        ipa_bias_kernel<<<(NBATCH * NRES * NRES) / 256, 256, 0, stream>>>(
            att, edge, P_wb + nb * CZd * Hd, P_gamma + nb * Hd, qpbuf, kpbuf, c_b, c_pt);
        softmax_kernel<<<NBATCH * Hd * NRES, 128, 0, stream>>>(att, resm, Hd * NRES);
        // o = a @ v  -> ocat[:, 0:2048]
        gemm(att, NRES, (ll)Hd * NRES * NRES, (ll)NRES * NRES,
             vbuf, 2048, (ll)NRES * 2048, CHd, 0, nullptr,
             ocat, 3456, (ll)NRES * 3456, CHd,
             NRES, CHd, NRES, NBATCH * Hd, Hd, 1.f, 0);
        // opt = a @ vp
        gemm(att, NRES, (ll)Hd * NRES * NRES, (ll)NRES * NRES,
             vpbuf, 288, (ll)NRES * 288, 36, 0, nullptr,
             optb, 288, (ll)NRES * 288, 36,
             NRES, 36, NRES, NBATCH * Hd, Hd, 1.f, 0);
        // opr[h,c] = sum_j a[b,h,i,j] edge[b,i,j,c]  (batched over (b,i), M=8)
        gemm(att, NRES * NRES, (ll)Hd * NRES * NRES, NRES,
             edge, CZd, (ll)NRES * NRES * CZd, (ll)NRES * CZd, 0, nullptr,
             ocat + 2432, CZd, (ll)NRES * 3456, 3456,
             Hd, CZd, NRES, ROWS, NRES, 1.f, 0);
        point_post_kernel<<<ROWS * 96 / 256, 256, 0, stream>>>(optb, Rbuf, tbuf, ocat);
        gemm(ocat, 3456, 0, 0, P_wo + (ll)nb * 3456 * CSd, CSd, 0, 0, 0, nullptr,
             thb, CSd, 0, 0, ROWS, CSd, 3456, 1, 1, 1.f, 0);
        add_mask_kernel<<<ROWS, 256, 0, stream>>>(sbuf, thb, resm);

        // ---------------- transformer (2 layers) ----------------
        hipMemcpyAsync(xxbuf, sbuf, (size_t)ROWS * CSd * sizeof(float),
                       hipMemcpyDeviceToDevice, stream);
        for (int l = 0; l < 2; ++l) {
            int bl = nb * 2 + l;
            ln_kernel<<<ROWS, 256, 0, stream>>>(xxbuf, thb, P_tln1g + bl * CSd, P_tln1b + bl * CSd);
            gemm(thb, CSd, 0, 0, P_wqkv + (ll)bl * CSd * 768, 768, 0, 0, 0, P_bqkv + bl * 768,
                 tqkv, 768, 0, 0, ROWS, 768, CSd, 1, 1, 1.f, 0);
            gemm(tqkv, 768, (ll)NRES * 768, 64, tqkv + 256, 768, (ll)NRES * 768, 64, 1, nullptr,
                 att2, NRES, (ll)THd * NRES * NRES, (ll)NRES * NRES,
                 NRES, NRES, 64, NBATCH * THd, THd, 0.125f, 0);
            softmax_kernel<<<NBATCH * THd * NRES, 128, 0, stream>>>(att2, resm, THd * NRES);
            gemm(att2, NRES, (ll)THd * NRES * NRES, (ll)NRES * NRES,
                 tqkv + 512, 768, (ll)NRES * 768, 64, 0, nullptr,
                 thb, CSd, (ll)NRES * CSd, 64,
                 NRES, 64, NRES, NBATCH * THd, THd, 1.f, 0);
            gemm(thb, CSd, 0, 0, P_wao + (ll)bl * CSd * CSd, CSd, 0, 0, 0, P_bao + bl * CSd,
                 tffb, CSd, 0, 0, ROWS, CSd, CSd, 1, 1, 1.f, 0);
            add_inplace_kernel<<<ROWS, 256, 0, stream>>>(xxbuf, tffb);
            ln_kernel<<<ROWS, 256, 0, stream>>>(xxbuf, thb, P_tln2g + bl * CSd, P_tln2b + bl * CSd);
            gemm(thb, CSd, 0, 0, P_w1 + (ll)bl * CSd * CSd, CSd, 0, 0, 0, P_b1 + bl * CSd,
                 tffb, CSd, 0, 0, ROWS, CSd, CSd, 1, 1, 1.f, 1);
            gemm(tffb, CSd, 0, 0, P_w2 + (ll)bl * CSd * CSd, CSd, 0, 0, 0, P_b2 + bl * CSd,
                 thb, CSd, 0, 0, ROWS, CSd, CSd, 1, 1, 1.f, 0);
            add_inplace_kernel<<<ROWS, 256, 0, stream>>>(xxbuf, thb);
        }
        gemm(xxbuf, CSd, 0, 0, P_postw + (ll)nb * CSd * CSd, CSd, 0, 0, 0, nullptr,
             thb, CSd, 0, 0, ROWS, CSd, CSd, 1, 1, 1.f, 0);
        add_mask_kernel<<<ROWS, 256, 0, stream>>>(sbuf, thb, resm);

        // ---------------- transition ----------------
        gemm(sbuf, CSd, 0, 0, P_trw1 + (ll)nb * CSd * CSd, CSd, 0, 0, 0, P_trb1 + nb * CSd,
             thb, CSd, 0, 0, ROWS, CSd, CSd, 1, 1, 1.f, 1);
        gemm(thb, CSd, 0, 0, P_trw2 + (ll)nb * CSd * CSd, CSd, 0, 0, 0, P_trb2 + nb * CSd,
             tffb, CSd, 0, 0, ROWS, CSd, CSd, 1, 1, 1.f, 1);
        gemm(tffb, CSd, 0, 0, P_trw3 + (ll)nb * CSd * CSd, CSd, 0, 0, 0, P_trb3 + nb * CSd,
             thb, CSd, 0, 0, ROWS, CSd, CSd, 1, 1, 1.f, 0);
        add_out_kernel<<<ROWS, 256, 0, stream>>>(tffb, sbuf, thb);
        ln_kernel<<<ROWS, 256, 0, stream>>>(tffb, sbuf, P_trlng + nb * CSd, P_trlnb + nb * CSd);

        // ---------------- backbone update ----------------
        small_gemm_kernel<<<(ROWS * 6 + 255) / 256, 256, 0, stream>>>(
            sbuf, P_bbw + (ll)nb * CSd * 6, P_bbb + nb * 6, dmask, updb, CSd, 6);
        rigid_update_kernel<<<3, 256, 0, stream>>>(updb, Rbuf, tbuf);
    }

    // ---------------- torsion head + output ----------------
    gemm(sbuf, CSd, 0, 0, P_tow1, CSd, 0, 0, 0, P_tob1, thb, CSd, 0, 0,
         ROWS, CSd, CSd, 1, 1, 1.f, 1);
    gemm(thb, CSd, 0, 0, P_tow2, CSd, 0, 0, 0, P_tob2, tffb, CSd, 0, 0,
         ROWS, CSd, CSd, 1, 1, 1.f, 0);
    add_inplace_kernel<<<ROWS, 256, 0, stream>>>(tffb, sbuf);
    small_gemm_kernel<<<(ROWS * 2 + 255) / 256, 256, 0, stream>>>(
        tffb, P_towf, P_tobf, nullptr, ubuf, CSd, 2);
    assemble_kernel<<<3, 256, 0, stream>>>(Rbuf, tbuf, ubuf, (float*)d_out);
}